// Valo_24953759990583
// MI455X (gfx1250) — compile-verified
//
#include <hip/hip_runtime.h>
#include <hip/hip_bf16.h>
#include <cfloat>
#include <cmath>
#include <cstdint>

#define NB   256
#define NP   256
#define NV   4096
#define KTOP 40

typedef float v4f __attribute__((ext_vector_type(4)));

// Low 32 bits of a generic pointer to a __shared__ object == wave-relative LDS
// byte offset (LDS aperture only occupies the upper 32 bits on gfx1250).
__device__ __forceinline__ uint32_t lds_off_u32(const void* p) {
    return (uint32_t)(uintptr_t)p;
}

// ---------------------------------------------------------------------------
// Kernel 1: max over patch axis.  logits[NB][NP][NV] -> maxlog[NB][NV]
// 1 GiB streamed once: use non-temporal b128 loads so the stream does not
// thrash the 192MB L2 (maxlog/token_probs then stay L2-resident).
// grid = NB * 4 (4 v-tiles of 1024 floats), block = 256 (8 waves, wave32).
// ---------------------------------------------------------------------------
__global__ __launch_bounds__(256) void maxred_kernel(const float* __restrict__ logits,
                                                     float* __restrict__ maxlog) {
    const int b     = blockIdx.x >> 2;
    const int vtile = blockIdx.x & 3;
    const int v4i   = vtile * 256 + threadIdx.x;           // float4 index in row

    const v4f* base = (const v4f*)(logits + (size_t)b * NP * NV) + v4i;

    v4f m;
    m.x = m.y = m.z = m.w = -FLT_MAX;
#pragma unroll 8
    for (int p = 0; p < NP; ++p) {
        v4f x = __builtin_nontemporal_load(base + (size_t)p * (NV / 4));
        m.x = fmaxf(m.x, x.x);
        m.y = fmaxf(m.y, x.y);
        m.z = fmaxf(m.z, x.z);
        m.w = fmaxf(m.w, x.w);
    }
    ((v4f*)(maxlog + (size_t)b * NV))[v4i] = m;
}

// ---------------------------------------------------------------------------
// Kernel 2: per-batch-row top-K, token_probs, weighted-BCE partial sum and
// top-k/audio intersection count.  One block (8 waves) per row.
// The 16KB row is staged into LDS with gfx1250 async global->LDS b128 copies.
// ---------------------------------------------------------------------------
__global__ __launch_bounds__(256) void topk_loss_kernel(const float* __restrict__ maxlog,
                                                        const int*   __restrict__ audio,
                                                        float*       __restrict__ probs_out,
                                                        float*       __restrict__ loss_part,
                                                        float*       __restrict__ acc_part) {
    __shared__ float         row[NV];       // 16KB   original max logits
    __shared__ float         work[NV];      // 16KB   destructible copy for argmax
    __shared__ float         rval[256];
    __shared__ int           ridx[256];
    __shared__ int           topk[KTOP];
    __shared__ int           satok[KTOP];
    __shared__ unsigned char selflag[NV];   // 4KB top-k membership
    __shared__ unsigned char tgtflag[NV];   // 4KB target membership
    __shared__ int           hitcnt;

    const int b   = blockIdx.x;
    const int tid = threadIdx.x;

    // ---- async copy of 16KB row: 4 x b128 per lane, tracked by ASYNCcnt ----
    {
        const uint64_t gbase = (uint64_t)(uintptr_t)(maxlog + (size_t)b * NV);
        const uint32_t lbase = lds_off_u32(row);
#pragma unroll
        for (int i = 0; i < 4; ++i) {
            uint32_t byteoff = (uint32_t)((i * 256 + tid) * 16);
            uint32_t laddr   = lbase + byteoff;
            asm volatile("global_load_async_to_lds_b128 %0, %1, %2 offset:0"
                         :: "v"(laddr), "v"(byteoff), "s"(gbase) : "memory");
        }
        asm volatile("s_wait_asynccnt 0x0" ::: "memory");
    }

    // zero flags, fetch audio tokens (after this wave's async copies have completed)
    for (int i = tid; i < NV; i += 256) { selflag[i] = 0; tgtflag[i] = 0; }
    if (tid < KTOP) satok[tid] = audio[b * KTOP + tid];
    if (tid == 0)   hitcnt = 0;
    __syncthreads();                        // row[] now visible to all waves

    for (int i = tid; i < NV; i += 256) work[i] = row[i];
    if (tid < KTOP) tgtflag[satok[tid]] = 1;
    __syncthreads();

    // ---- 40 rounds of block-wide argmax (tie-break: lowest index, like
    //      jax.lax.top_k).  Strided per-thread scan keeps ties in order. ----
    for (int k = 0; k < KTOP; ++k) {
        float bv = -FLT_MAX;
        int   bi = NV;
        for (int i = tid; i < NV; i += 256) {
            float v = work[i];
            if (v > bv) { bv = v; bi = i; }     // first (smallest) index wins ties
        }
        rval[tid] = bv; ridx[tid] = bi;
        __syncthreads();
        for (int s = 128; s > 0; s >>= 1) {
            if (tid < s) {
                float ov = rval[tid + s]; int oi = ridx[tid + s];
                float mv = rval[tid];     int mi = ridx[tid];
                if (ov > mv || (ov == mv && oi < mi)) { rval[tid] = ov; ridx[tid] = oi; }
            }
            __syncthreads();
        }
        if (tid == 0) {
            int sel = ridx[0];
            topk[k]      = sel;
            work[sel]    = -FLT_MAX;
            selflag[sel] = 1;
        }
        __syncthreads();
    }

    // ---- token_probs + weighted clamped BCE partial sum ----
    float lsum = 0.0f;
    for (int i = tid; i < NV; i += 256) {
        float x   = row[i];
        bool  sel = selflag[i] != 0;
        float p   = sel ? (1.0f / (1.0f + expf(-x))) : 0.0f;
        probs_out[(size_t)b * NV + i] = p;

        bool  tgt    = tgtflag[i] != 0;
        float t      = tgt ? 1.0f : 0.0f;
        float w      = tgt ? 50.0f : 1.0f;
        float logp   = fmaxf(logf(p),     -100.0f);   // log(0) = -inf -> -100
        float log1mp = fmaxf(log1pf(-p),  -100.0f);   // p==1  -> -inf -> -100
        float bce    = -(t * logp + (1.0f - t) * log1mp);
        lsum += w * bce;
    }
    rval[tid] = lsum;
    __syncthreads();
    for (int s = 128; s > 0; s >>= 1) {
        if (tid < s) rval[tid] += rval[tid + s];
        __syncthreads();
    }

    // ---- |topk ∩ audio| (integer LDS atomic: order-independent) ----
    if (tid < KTOP) {
        int my = topk[tid];
        int hit = 0;
        for (int j = 0; j < KTOP; ++j) hit |= (my == satok[j]);
        if (hit) atomicAdd(&hitcnt, 1);
    }
    __syncthreads();
    if (tid == 0) {
        loss_part[b] = rval[0];
        acc_part[b]  = (float)hitcnt;
    }
}

// ---------------------------------------------------------------------------
// Kernel 3: deterministic final reduction of 256 partials into the 2 scalars.
// ---------------------------------------------------------------------------
__global__ __launch_bounds__(256) void finalize_kernel(const float* __restrict__ loss_part,
                                                       const float* __restrict__ acc_part,
                                                       float* __restrict__ out_loss,
                                                       float* __restrict__ out_acc) {
    __shared__ float l[256];
    __shared__ float a[256];
    const int tid = threadIdx.x;
    l[tid] = loss_part[tid];
    a[tid] = acc_part[tid];
    __syncthreads();
    for (int s = 128; s > 0; s >>= 1) {
        if (tid < s) { l[tid] += l[tid + s]; a[tid] += a[tid + s]; }
        __syncthreads();
    }
    if (tid == 0) {
        *out_loss = l[0] / ((float)NB * (float)NV);
        *out_acc  = a[0] / ((float)KTOP * (float)NB);
    }
}

extern "C" void kernel_launch(void* const* d_in, const int* in_sizes, int n_in,
                              void* d_out, int out_size, void* d_ws, size_t ws_size,
                              hipStream_t stream) {
    (void)in_sizes; (void)n_in; (void)out_size; (void)ws_size;

    const float* logits = (const float*)d_in[0];   // [256,256,4096] f32
    const int*   audio  = (const int*)d_in[1];     // [256,40] i32
    float*       out    = (float*)d_out;           // [loss | probs(1M) | acc]

    float* maxlog    = (float*)d_ws;               // 4MB
    float* loss_part = maxlog + (size_t)NB * NV;   // 256 floats
    float* acc_part  = loss_part + NB;             // 256 floats

    maxred_kernel<<<NB * 4, 256, 0, stream>>>(logits, maxlog);
    topk_loss_kernel<<<NB, 256, 0, stream>>>(maxlog, audio, out + 1, loss_part, acc_part);
    finalize_kernel<<<1, 256, 0, stream>>>(loss_part, acc_part,
                                           out, out + 1 + (size_t)NB * NV);
}